// MCAM_39505109189210
// MI455X (gfx1250) — compile-verified
//
#include <hip/hip_runtime.h>

typedef __attribute__((ext_vector_type(2))) float v2f;
typedef __attribute__((ext_vector_type(8))) float v8f;

#define HW    16384
#define CHW   (128*HW)
#define NPOS  131072        /* B*HW */
#define NTOT  16777216L     /* B*C*HW */

static __device__ __forceinline__ v8f wmma_f32(v2f a, v2f b, v8f c) {
  return __builtin_amdgcn_wmma_f32_16x16x4_f32(false, a, false, b, (short)0, c, false, false);
}
static __device__ __forceinline__ v8f vzero8() { v8f z; for (int i = 0; i < 8; i++) z[i] = 0.f; return z; }

// ---------------------------------------------------------------- LayerNorm over C
__global__ void k_ln(const float* __restrict__ x, const float* __restrict__ lw,
                     const float* __restrict__ lb, float* __restrict__ y) {
  long p = (long)blockIdx.x * 256 + threadIdx.x;   // 131072 positions
  long b = p >> 14, hw = p & 16383;
  const float* xp = x + b * CHW + hw;
  float s = 0.f, s2 = 0.f;
  for (int c = 0; c < 128; c++) { float v = xp[(long)c * HW]; s += v; s2 += v * v; }
  float mu = s * (1.f / 128.f);
  float var = s2 * (1.f / 128.f) - mu * mu;
  float inv = rsqrtf(var + 1e-5f);
  float* yp = y + b * CHW + hw;
  for (int c = 0; c < 128; c++) {
    float v = xp[(long)c * HW];
    yp[(long)c * HW] = (v - mu) * inv * lw[c] + lb[c];
  }
}

// ---------------------------------------------------------------- 1x1 dense conv (WMMA), M=outch N=pos
template<bool ADD_RES>
__global__ void k_conv1x1(const float* __restrict__ in, const float* __restrict__ wt,
                          const float* __restrict__ res, float* __restrict__ out) {
  const int ptile = blockIdx.x;            // 4096 tiles of 32 positions
  const int o0 = blockIdx.y * 64;          // 2 out-channel tiles of 64
  const long pbase = (long)ptile * 32;
  const long b = pbase >> 14, hw = pbase & 16383;
  __shared__ float Xs[32][33];             // [cin][pos]
  __shared__ float Ws[64][33];             // [outch][cin]
  const int tid = threadIdx.x, lane = tid & 31, wv = tid >> 5;
  const int l16 = lane & 15, hi = lane >> 4;
  const int msub = wv >> 1, nsub = wv & 1; // 4 och-subtiles x 2 pos-subtiles
  v8f acc = vzero8();
  for (int c0 = 0; c0 < 128; c0 += 32) {
    for (int i = tid; i < 1024; i += 256) {
      int c = i >> 5, pp = i & 31;
      Xs[c][pp] = in[b * CHW + (long)(c0 + c) * HW + hw + pp];
    }
    for (int i = tid; i < 2048; i += 256) {
      int o = i >> 5, c = i & 31;
      Ws[o][c] = wt[(long)(o0 + o) * 128 + c0 + c];
    }
    __syncthreads();
    for (int k0 = 0; k0 < 32; k0 += 4) {
      v2f a, bb;
      a.x = Ws[msub * 16 + l16][k0 + 2 * hi];
      a.y = Ws[msub * 16 + l16][k0 + 2 * hi + 1];
      bb.x = Xs[k0 + 2 * hi][nsub * 16 + l16];
      bb.y = Xs[k0 + 2 * hi + 1][nsub * 16 + l16];
      acc = wmma_f32(a, bb, acc);
    }
    __syncthreads();
  }
  for (int r = 0; r < 8; r++) {
    int o = o0 + msub * 16 + r + 8 * hi;
    long idx = b * CHW + (long)o * HW + hw + nsub * 16 + l16;
    float v = acc[r];
    if (ADD_RES) v += res[idx];
    out[idx] = v;
  }
}

// ---------------------------------------------------------------- dilated dense 3x3 conv (WMMA, 9 taps)
template<int DIL>
__global__ void k_conv3x3(const float* __restrict__ in, const float* __restrict__ wt,
                          float* __restrict__ out) {
  const int wti = blockIdx.x & 3, otile = blockIdx.x >> 2;
  const int h = blockIdx.y, b = blockIdx.z;
  const int w0 = wti * 32, o0 = otile * 64;
  const int WT = 32 + 2 * DIL;
  __shared__ float Xs[16][3][64];     // [cin][ty][w-with-halo]
  __shared__ float Wls[64][16][9];    // [outch][cin][tap]
  const int tid = threadIdx.x, lane = tid & 31, wv = tid >> 5;
  const int l16 = lane & 15, hi = lane >> 4;
  const int msub = wv >> 1, nsub = wv & 1;
  v8f acc = vzero8();
  for (int c0 = 0; c0 < 128; c0 += 16) {
    for (int i = tid; i < 16 * 3 * WT; i += 256) {
      int c = i / (3 * WT), rem = i % (3 * WT), ty = rem / WT, ix = rem % WT;
      int hh = h + (ty - 1) * DIL, ww = w0 - DIL + ix;
      float v = 0.f;
      if (hh >= 0 && hh < 128 && ww >= 0 && ww < 128)
        v = in[((long)b * 128 + c0 + c) * HW + hh * 128 + ww];
      Xs[c][ty][ix] = v;
    }
    for (int i = tid; i < 9216; i += 256) {
      int o = i / 144, rem = i % 144, c = rem / 9, tp = rem % 9;
      Wls[o][c][tp] = wt[((long)(o0 + o) * 128 + c0 + c) * 9 + tp];
    }
    __syncthreads();
    for (int k0 = 0; k0 < 16; k0 += 4) {
      for (int ty = 0; ty < 3; ty++) {
        for (int tx = 0; tx < 3; tx++) {
          v2f a, bb;
          a.x = Wls[msub * 16 + l16][k0 + 2 * hi][ty * 3 + tx];
          a.y = Wls[msub * 16 + l16][k0 + 2 * hi + 1][ty * 3 + tx];
          bb.x = Xs[k0 + 2 * hi][ty][nsub * 16 + l16 + tx * DIL];
          bb.y = Xs[k0 + 2 * hi + 1][ty][nsub * 16 + l16 + tx * DIL];
          acc = wmma_f32(a, bb, acc);
        }
      }
    }
    __syncthreads();
  }
  for (int r = 0; r < 8; r++) {
    int o = o0 + msub * 16 + r + 8 * hi;
    out[((long)b * 128 + o) * HW + h * 128 + w0 + nsub * 16 + l16] = acc[r];
  }
}

// ---------------------------------------------------------------- BN stats
__global__ void k_zero(float* p, int n) {
  for (int i = threadIdx.x; i < n; i += 256) p[i] = 0.f;
}
__global__ void k_bnstats(const float* __restrict__ y, float* __restrict__ st) {
  const int c = blockIdx.x, slab = blockIdx.y, tid = threadIdx.x;
  long base = (long)slab * 8192 + tid;
  float s = 0.f, s2 = 0.f;
  for (int k = 0; k < 32; k++) {
    long p = base + (long)k * 256;
    long b = p >> 14, hw = p & 16383;
    float v = y[(b * 128 + c) * (long)HW + hw];
    s += v; s2 += v * v;
  }
  __shared__ float sh[256], sh2[256];
  sh[tid] = s; sh2[tid] = s2; __syncthreads();
  for (int off = 128; off > 0; off >>= 1) {
    if (tid < off) { sh[tid] += sh[tid + off]; sh2[tid] += sh2[tid + off]; }
    __syncthreads();
  }
  if (tid == 0) { atomicAdd(&st[c], sh[0]); atomicAdd(&st[128 + c], sh2[0]); }
}
__global__ void k_bnfin(const float* __restrict__ st,
                        const float* g1, const float* be1, const float* g2, const float* be2,
                        const float* g3, const float* be3, float* __restrict__ ss) {
  int c = threadIdx.x;
  const float* gs[3] = {g1, g2, g3};
  const float* bs[3] = {be1, be2, be3};
  const float invN = 1.f / 131072.f;
  for (int br = 0; br < 3; br++) {
    float mu = st[br * 256 + c] * invN;
    float var = st[br * 256 + 128 + c] * invN - mu * mu;
    float sc = gs[br][c] * rsqrtf(var + 1e-5f);
    ss[br * 256 + c] = sc;
    ss[br * 256 + 128 + c] = bs[br][c] - mu * sc;
  }
}
__global__ void k_bnrelusum(const float* __restrict__ y1, const float* __restrict__ y2,
                            const float* __restrict__ y3, const float* __restrict__ ss,
                            float* __restrict__ s) {
  long i = (long)blockIdx.x * 256 + threadIdx.x;
  int c = (int)((i >> 14) & 127);
  float a = fmaxf(y1[i] * ss[c] + ss[128 + c], 0.f);
  float b = fmaxf(y2[i] * ss[256 + c] + ss[384 + c], 0.f);
  float d = fmaxf(y3[i] * ss[512 + c] + ss[640 + c], 0.f);
  s[i] = a + b + d;
}

// ---------------------------------------------------------------- six depthwise convs summed
__global__ void k_dwatt(const float* __restrict__ x1,
                        const float* w1, const float* b1, const float* w2, const float* b2,
                        const float* w3, const float* b3, const float* w4, const float* b4,
                        const float* w5, const float* b5, const float* w6, const float* b6,
                        float* __restrict__ att) {
  const int wti = blockIdx.x & 3, hti = blockIdx.x >> 2;
  const int c = blockIdx.y, b = blockIdx.z;
  const int w0 = wti * 32, h0 = hti * 8;
  __shared__ float T[14][40];
  __shared__ float wsm[36];
  const int tid = threadIdx.x;
  const float* base = x1 + ((long)b * 128 + c) * HW;
  for (int i = tid; i < 14 * 38; i += 256) {
    int r = i / 38, cc = i % 38;
    int hh = h0 - 3 + r, ww = w0 - 3 + cc;
    float v = 0.f;
    if (hh >= 0 && hh < 128 && ww >= 0 && ww < 128) v = base[hh * 128 + ww];
    T[r][cc] = v;
  }
  if (tid < 5) wsm[tid] = w1[(long)c * 5 + tid];
  else if (tid < 10) wsm[tid] = w2[(long)c * 5 + tid - 5];
  else if (tid < 11) wsm[tid] = w3[c];
  else if (tid < 20) wsm[tid] = w4[(long)c * 9 + tid - 11];
  else if (tid < 27) wsm[tid] = w5[(long)c * 7 + tid - 20];
  else if (tid < 34) wsm[tid] = w6[(long)c * 7 + tid - 27];
  else if (tid == 34) wsm[34] = b1[c] + b2[c] + b3[c] + b4[c] + b5[c] + b6[c];
  __syncthreads();
  int ww = tid & 31, hh = tid >> 5;
  int R = hh + 3, Cx = ww + 3;
  float acc = wsm[34];
  for (int d = -2; d <= 2; d++) acc += T[R][Cx + d] * wsm[d + 2];
  for (int d = -2; d <= 2; d++) acc += T[R + d][Cx] * wsm[5 + d + 2];
  acc += T[R][Cx] * wsm[10];
  for (int dy = -1; dy <= 1; dy++)
    for (int dx = -1; dx <= 1; dx++) acc += T[R + dy][Cx + dx] * wsm[11 + (dy + 1) * 3 + dx + 1];
  for (int d = -3; d <= 3; d++) acc += T[R][Cx + d] * wsm[20 + d + 3];
  for (int d = -3; d <= 3; d++) acc += T[R + d][Cx] * wsm[27 + d + 3];
  att[((long)b * 128 + c) * HW + (h0 + hh) * 128 + w0 + ww] = acc;
}

// ---------------------------------------------------------------- l2 norms (q1==k2 / q2 / k1)
__global__ void k_norms(const float* __restrict__ out1, const float* __restrict__ out2,
                        float* __restrict__ nq1, float* __restrict__ nq2, float* __restrict__ nk1) {
  int bh = blockIdx.x, b = bh >> 3, hd = bh & 7, i = threadIdx.x;
  const float* O1 = out1 + ((long)b * 128 + hd * 16) * HW;
  const float* O2 = out2 + ((long)b * 128 + hd * 16) * HW;
  float c1 = 0.f, c2 = 0.f, r1 = 0.f;
  for (int c = 0; c < 16; c++) {
    const float* p1 = O1 + (long)c * HW;
    const float* p2 = O2 + (long)c * HW;
    for (int p = 0; p < 128; p++) {
      float v2 = p2[p * 128 + i]; c2 += v2 * v2;
      float v1 = p1[p * 128 + i]; c1 += v1 * v1;
      float u1 = p1[i * 128 + p]; r1 += u1 * u1;
    }
  }
  nq1[bh * 128 + i] = fmaxf(sqrtf(c2), 1e-12f);
  nq2[bh * 128 + i] = fmaxf(sqrtf(c1), 1e-12f);
  nk1[bh * 128 + i] = fmaxf(sqrtf(r1), 1e-12f);
}

// ---------------------------------------------------------------- Gram matrices, K=2048 (WMMA)
// TRANSA=false: M1[t,i] = sum_{c,p} O1[t,p] O2[p,i]
// TRANSA=true : M2[i,t] = sum_{c,p} O1[p,i] O2[p,t]
template<bool TRANSA>
__global__ void k_gram(const float* __restrict__ out1, const float* __restrict__ out2,
                       float* __restrict__ M) {
  int bh = blockIdx.x, b = bh >> 3, hd = bh & 7;
  const float* O1 = out1 + ((long)b * 128 + hd * 16) * HW;
  const float* O2 = out2 + ((long)b * 128 + hd * 16) * HW;
  __shared__ float As[16][136], Bs[16][136];
  const int tid = threadIdx.x, lane = tid & 31, wv = tid >> 5;
  const int l16 = lane & 15, hi = lane >> 4;
  v8f acc[8];
  for (int t = 0; t < 8; t++) acc[t] = vzero8();
  for (int c = 0; c < 16; c++) {
    const float* P1 = O1 + (long)c * HW;
    const float* P2 = O2 + (long)c * HW;
    for (int p0 = 0; p0 < 128; p0 += 16) {
      for (int i = tid; i < 2048; i += 256) {
        int k = i >> 7, n = i & 127;
        Bs[k][n] = P2[(p0 + k) * 128 + n];
      }
      if (TRANSA) {
        for (int i = tid; i < 2048; i += 256) {
          int k = i >> 7, n = i & 127;
          As[k][n] = P1[(p0 + k) * 128 + n];
        }
      } else {
        for (int i = tid; i < 2048; i += 256) {
          int k = i & 15, t = i >> 4;
          As[k][t] = P1[t * 128 + p0 + k];
        }
      }
      __syncthreads();
      for (int k0 = 0; k0 < 16; k0 += 4) {
        v2f a;
        a.x = As[k0 + 2 * hi][wv * 16 + l16];
        a.y = As[k0 + 2 * hi + 1][wv * 16 + l16];
        for (int ct = 0; ct < 8; ct++) {
          v2f bb;
          bb.x = Bs[k0 + 2 * hi][ct * 16 + l16];
          bb.y = Bs[k0 + 2 * hi + 1][ct * 16 + l16];
          acc[ct] = wmma_f32(a, bb, acc[ct]);
        }
      }
      __syncthreads();
    }
  }
  float* Mp = M + (long)bh * 16384;
  for (int ct = 0; ct < 8; ct++)
    for (int r = 0; r < 8; r++)
      Mp[(wv * 16 + r + 8 * hi) * 128 + ct * 16 + l16] = acc[ct][r];
}

// ---------------------------------------------------------------- softmax rows of scaled M1^T / M2
__global__ void k_softmax(const float* __restrict__ M1, const float* __restrict__ M2,
                          const float* __restrict__ nq1, const float* __restrict__ nq2,
                          const float* __restrict__ nk1,
                          float* __restrict__ A1, float* __restrict__ A2) {
  int bh = blockIdx.x, i = threadIdx.x;
  const float* m1 = M1 + (long)bh * 16384;
  const float* m2 = M2 + (long)bh * 16384;
  const float* q1n = nq1 + bh * 128;
  const float* q2n = nq2 + bh * 128;
  const float* k1n = nk1 + bh * 128;
  float* a1 = A1 + (long)bh * 16384 + (long)i * 128;
  float* a2 = A2 + (long)bh * 16384 + (long)i * 128;
  float inv1 = 1.f / q1n[i], inv2 = 1.f / q2n[i];
  float mx1 = -3.4e38f, mx2 = -3.4e38f;
  for (int t = 0; t < 128; t++) {
    float s1 = m1[t * 128 + i] * inv1 / k1n[t];
    float s2 = m2[i * 128 + t] * inv2 / q1n[t];
    mx1 = fmaxf(mx1, s1); mx2 = fmaxf(mx2, s2);
  }
  float sm1 = 0.f, sm2 = 0.f;
  for (int t = 0; t < 128; t++) {
    float s1 = m1[t * 128 + i] * inv1 / k1n[t];
    float s2 = m2[i * 128 + t] * inv2 / q1n[t];
    float e1 = __expf(s1 - mx1), e2 = __expf(s2 - mx2);
    sm1 += e1; sm2 += e2;
    a1[t] = e1; a2[t] = e2;
  }
  float r1 = 1.f / sm1, r2 = 1.f / sm2;
  for (int t = 0; t < 128; t++) { a1[t] *= r1; a2[t] *= r2; }
}

// ---------------------------------------------------------------- o34 = A1*O1_c + O2_c*A2^T + residual terms
__global__ void k_apply(const float* __restrict__ out1, const float* __restrict__ out2,
                        const float* __restrict__ A1, const float* __restrict__ A2,
                        const float* __restrict__ nq1, const float* __restrict__ nq2,
                        float* __restrict__ o34) {
  int bc = blockIdx.x, b = bc >> 7, ch = bc & 127, hd = ch >> 4, bh = b * 8 + hd;
  const float* O1 = out1 + ((long)b * 128 + ch) * HW;
  const float* O2 = out2 + ((long)b * 128 + ch) * HW;
  const float* A1p = A1 + (long)bh * 16384;
  const float* A2p = A2 + (long)bh * 16384;
  const float* q1n = nq1 + bh * 128;
  const float* q2n = nq2 + bh * 128;
  __shared__ float As[16][136], Bs[16][136];
  const int tid = threadIdx.x, lane = tid & 31, wv = tid >> 5;
  const int l16 = lane & 15, hi = lane >> 4;
  v8f acc[8];
  for (int t = 0; t < 8; t++) acc[t] = vzero8();
  // GEMM1: A1 (h x t)  *  O1 (t x w)
  for (int t0 = 0; t0 < 128; t0 += 16) {
    for (int i = tid; i < 2048; i += 256) {
      int k = i & 15, hrow = i >> 4;
      As[k][hrow] = A1p[hrow * 128 + t0 + k];
    }
    for (int i = tid; i < 2048; i += 256) {
      int k = i >> 7, n = i & 127;
      Bs[k][n] = O1[(t0 + k) * 128 + n];
    }
    __syncthreads();
    for (int k0 = 0; k0 < 16; k0 += 4) {
      v2f a;
      a.x = As[k0 + 2 * hi][wv * 16 + l16];
      a.y = As[k0 + 2 * hi + 1][wv * 16 + l16];
      for (int ct = 0; ct < 8; ct++) {
        v2f bb;
        bb.x = Bs[k0 + 2 * hi][ct * 16 + l16];
        bb.y = Bs[k0 + 2 * hi + 1][ct * 16 + l16];
        acc[ct] = wmma_f32(a, bb, acc[ct]);
      }
    }
    __syncthreads();
  }
  // GEMM2: O2 (h x t)  *  A2^T (t x w)
  for (int t0 = 0; t0 < 128; t0 += 16) {
    for (int i = tid; i < 2048; i += 256) {
      int k = i & 15, hrow = i >> 4;
      As[k][hrow] = O2[hrow * 128 + t0 + k];
    }
    for (int i = tid; i < 2048; i += 256) {
      int k = i & 15, wcol = i >> 4;
      Bs[k][wcol] = A2p[wcol * 128 + t0 + k];
    }
    __syncthreads();
    for (int k0 = 0; k0 < 16; k0 += 4) {
      v2f a;
      a.x = As[k0 + 2 * hi][wv * 16 + l16];
      a.y = As[k0 + 2 * hi + 1][wv * 16 + l16];
      for (int ct = 0; ct < 8; ct++) {
        v2f bb;
        bb.x = Bs[k0 + 2 * hi][ct * 16 + l16];
        bb.y = Bs[k0 + 2 * hi + 1][ct * 16 + l16];
        acc[ct] = wmma_f32(a, bb, acc[ct]);
      }
    }
    __syncthreads();
  }
  float* Op = o34 + ((long)b * 128 + ch) * HW;
  for (int ct = 0; ct < 8; ct++) {
    for (int r = 0; r < 8; r++) {
      int h = wv * 16 + r + 8 * hi;
      int w = ct * 16 + l16;
      float v = acc[ct][r] + O2[w * 128 + h] / q1n[h] + O1[h * 128 + w] / q2n[w];
      Op[h * 128 + w] = v;
    }
  }
}

// ---------------------------------------------------------------- launch
extern "C" void kernel_launch(void* const* d_in, const int* in_sizes, int n_in,
                              void* d_out, int out_size, void* d_ws, size_t ws_size,
                              hipStream_t stream) {
  const float* x     = (const float*)d_in[0];
  const float* ln_w  = (const float*)d_in[1];
  const float* ln_b  = (const float*)d_in[2];
  const float* wd1   = (const float*)d_in[3];
  const float* gd1   = (const float*)d_in[4];
  const float* bd1   = (const float*)d_in[5];
  const float* wd2   = (const float*)d_in[6];
  const float* gd2   = (const float*)d_in[7];
  const float* bd2   = (const float*)d_in[8];
  const float* wd3   = (const float*)d_in[9];
  const float* gd3   = (const float*)d_in[10];
  const float* bd3   = (const float*)d_in[11];
  const float* w_o1  = (const float*)d_in[12];
  const float* w1    = (const float*)d_in[13];
  const float* b1    = (const float*)d_in[14];
  const float* w2    = (const float*)d_in[15];
  const float* b2    = (const float*)d_in[16];
  const float* w3    = (const float*)d_in[17];
  const float* b3    = (const float*)d_in[18];
  const float* w4    = (const float*)d_in[19];
  const float* b4    = (const float*)d_in[20];
  const float* w5    = (const float*)d_in[21];
  const float* b5    = (const float*)d_in[22];
  const float* w6    = (const float*)d_in[23];
  const float* b6    = (const float*)d_in[24];
  const float* w_o2  = (const float*)d_in[25];
  float* out = (float*)d_out;

  float* ws = (float*)d_ws;
  const size_t NB = 16777216ULL;
  float* BIG0 = ws;             // y1 -> S -> out1
  float* BIG1 = ws + NB;        // y2 -> out2
  float* BIG2 = ws + 2 * NB;    // y3 -> x1
  float* BIG3 = ws + 3 * NB;    // att -> o34
  float* STATS = ws + 4 * NB;           // 768
  float* SS    = STATS + 768;           // 768
  float* M1    = SS + 768;              // 64*128*128
  float* M2    = M1 + 1048576;
  float* A1    = M2 + 1048576;
  float* A2    = A1 + 1048576;
  float* NQ1   = A2 + 1048576;          // 64*128 each
  float* NQ2   = NQ1 + 8192;
  float* NK1   = NQ2 + 8192;

  // dense branch
  k_conv1x1<false><<<dim3(4096, 2), 256, 0, stream>>>(x, wd1, x, BIG0);
  k_conv3x3<6><<<dim3(8, 128, 8), 256, 0, stream>>>(x, wd2, BIG1);
  k_conv3x3<12><<<dim3(8, 128, 8), 256, 0, stream>>>(x, wd3, BIG2);
  k_zero<<<1, 256, 0, stream>>>(STATS, 768);
  k_bnstats<<<dim3(128, 16), 256, 0, stream>>>(BIG0, STATS);
  k_bnstats<<<dim3(128, 16), 256, 0, stream>>>(BIG1, STATS + 256);
  k_bnstats<<<dim3(128, 16), 256, 0, stream>>>(BIG2, STATS + 512);
  k_bnfin<<<1, 128, 0, stream>>>(STATS, gd1, bd1, gd2, bd2, gd3, bd3, SS);
  k_bnrelusum<<<65536, 256, 0, stream>>>(BIG0, BIG1, BIG2, SS, BIG0);
  k_conv1x1<true><<<dim3(4096, 2), 256, 0, stream>>>(BIG0, w_o1, x, BIG1);   // out2

  // attention input branch
  k_ln<<<512, 256, 0, stream>>>(x, ln_w, ln_b, BIG2);                        // x1
  k_dwatt<<<dim3(64, 128, 8), 256, 0, stream>>>(BIG2, w1, b1, w2, b2, w3, b3,
                                                w4, b4, w5, b5, w6, b6, BIG3);
  k_conv1x1<false><<<dim3(4096, 2), 256, 0, stream>>>(BIG3, w_o2, x, BIG0);  // out1

  // attention
  k_norms<<<64, 128, 0, stream>>>(BIG0, BIG1, NQ1, NQ2, NK1);
  k_gram<false><<<64, 256, 0, stream>>>(BIG0, BIG1, M1);
  k_gram<true><<<64, 256, 0, stream>>>(BIG0, BIG1, M2);
  k_softmax<<<64, 128, 0, stream>>>(M1, M2, NQ1, NQ2, NK1, A1, A2);
  k_apply<<<1024, 256, 0, stream>>>(BIG0, BIG1, A1, A2, NQ1, NQ2, BIG3);     // o3+o4

  // final fused conv: conv(o3,w)+conv(o4,w)+x == conv(o3+o4,w)+x
  k_conv1x1<true><<<dim3(4096, 2), 256, 0, stream>>>(BIG3, w_o2, x, out);
}